// RecurrentGCN_52656299049582
// MI455X (gfx1250) — compile-verified
//
#include <hip/hip_runtime.h>

#define F 64

typedef __attribute__((ext_vector_type(2))) float v2f;
typedef __attribute__((ext_vector_type(8))) float v8f;

// ---------------- utility ----------------
__global__ void zero_f32(float* __restrict__ p, long n) {
    long i = (long)blockIdx.x * blockDim.x + threadIdx.x;
    long stride = (long)gridDim.x * blockDim.x;
    for (; i < n; i += stride) p[i] = 0.0f;
}

// deg[row[e]] += w[e]
__global__ void deg_kernel(const int* __restrict__ row, const float* __restrict__ w,
                           float* __restrict__ deg, int E) {
    int e = blockIdx.x * blockDim.x + threadIdx.x;
    if (e < E) atomicAdd(&deg[row[e]], w[e]);
}

// norm[e] = -dis[row]*w*dis[col], dis = deg>0 ? rsqrt(deg) : 0
__global__ void norm_kernel(const int* __restrict__ row, const int* __restrict__ col,
                            const float* __restrict__ w, const float* __restrict__ deg,
                            float* __restrict__ nrm, int E) {
    int e = blockIdx.x * blockDim.x + threadIdx.x;
    if (e >= E) return;
    float dr = deg[row[e]], dc = deg[col[e]];
    float ir = dr > 0.f ? rsqrtf(dr) : 0.f;
    float ic = dc > 0.f ? rsqrtf(dc) : 0.f;
    nrm[e] = -ir * w[e] * ic;
}

// dst[col[e]] += norm[e] * src[row[e]]  (one wave32 per edge, 2 floats/lane)
__global__ void prop_kernel(const float* __restrict__ src, float* __restrict__ dst,
                            const int* __restrict__ row, const int* __restrict__ col,
                            const float* __restrict__ nrm, int E) {
    int wave = blockIdx.x * (blockDim.x >> 5) + (threadIdx.x >> 5);
    int lane = threadIdx.x & 31;
    if (wave >= E) return;
    int r = row[wave], c = col[wave];
    float nv = nrm[wave];
    const float2 xs = *(const float2*)(src + (long)r * F + lane * 2);
    float* dp = dst + (long)c * F + lane * 2;
    atomicAdd(dp + 0, nv * xs.x);
    atomicAdd(dp + 1, nv * xs.y);
}

// ---------------- WMMA gate GEMM ----------------
// Per wave: 16-row tile. acc(16x64) = x@W0 + tx1@W1 + (2*p2-x)@W2, for z and h
// weights staged in LDS (48KB per phase), V_WMMA_F32_16X16X4_F32, K stepped by 4.
__launch_bounds__(256)
__global__ void gemm_gate_kernel(const float* __restrict__ x,
                                 const float* __restrict__ tx1,
                                 const float* __restrict__ p2,
                                 const float* __restrict__ Wz,   // [3][64][64]
                                 const float* __restrict__ Wh,   // [3][64][64]
                                 const float* __restrict__ bxz, const float* __restrict__ bhz,
                                 const float* __restrict__ bxh, const float* __restrict__ bhh,
                                 float* __restrict__ hout, int n) {
    __shared__ float lw[3 * F * F];  // 48 KB
    const int tid = threadIdx.x;
    const int waveId = tid >> 5;
    const int lane = tid & 31;
    const int m  = lane & 15;   // A row / B col / C col within tile
    const int kh = lane >> 4;   // K half-select per ISA f32 A/B layout
    const int baseRow = (blockIdx.x * 8 + waveId) * 16;
    const bool valid = baseRow < n;
    const int r0 = valid ? baseRow : 0;

    const float* xr  = x   + (long)r0 * F;
    const float* t1r = tx1 + (long)r0 * F;
    const float* p2r = p2  + (long)r0 * F;

    v8f accz[4] = {};
    v8f acch[4] = {};

    // ---------- phase Z ----------
    for (int i = tid; i < 3 * F * F / 4; i += 256)
        ((float4*)lw)[i] = ((const float4*)Wz)[i];
    __syncthreads();

    for (int k = 0; k < 16; ++k) {
        const int kk = 4 * k + 2 * kh;
        float2 fx = *(const float2*)(xr  + m * F + kk);
        float2 f1 = *(const float2*)(t1r + m * F + kk);
        float2 fp = *(const float2*)(p2r + m * F + kk);
        v2f ax = {fx.x, fx.y};
        v2f a1 = {f1.x, f1.y};
        v2f a2 = {2.f * fp.x - fx.x, 2.f * fp.y - fx.y};
#pragma unroll
        for (int t = 0; t < 4; ++t) {
            const int nc = t * 16 + m;
            v2f b0 = {lw[0 * F * F + kk * F + nc], lw[0 * F * F + (kk + 1) * F + nc]};
            v2f b1 = {lw[1 * F * F + kk * F + nc], lw[1 * F * F + (kk + 1) * F + nc]};
            v2f b2 = {lw[2 * F * F + kk * F + nc], lw[2 * F * F + (kk + 1) * F + nc]};
            accz[t] = __builtin_amdgcn_wmma_f32_16x16x4_f32(false, ax, false, b0, (short)0, accz[t], false, false);
            accz[t] = __builtin_amdgcn_wmma_f32_16x16x4_f32(false, a1, false, b1, (short)0, accz[t], false, false);
            accz[t] = __builtin_amdgcn_wmma_f32_16x16x4_f32(false, a2, false, b2, (short)0, accz[t], false, false);
        }
    }
    __syncthreads();

    // ---------- phase H ----------
    for (int i = tid; i < 3 * F * F / 4; i += 256)
        ((float4*)lw)[i] = ((const float4*)Wh)[i];
    __syncthreads();

    for (int k = 0; k < 16; ++k) {
        const int kk = 4 * k + 2 * kh;
        float2 fx = *(const float2*)(xr  + m * F + kk);
        float2 f1 = *(const float2*)(t1r + m * F + kk);
        float2 fp = *(const float2*)(p2r + m * F + kk);
        v2f ax = {fx.x, fx.y};
        v2f a1 = {f1.x, f1.y};
        v2f a2 = {2.f * fp.x - fx.x, 2.f * fp.y - fx.y};
#pragma unroll
        for (int t = 0; t < 4; ++t) {
            const int nc = t * 16 + m;
            v2f b0 = {lw[0 * F * F + kk * F + nc], lw[0 * F * F + (kk + 1) * F + nc]};
            v2f b1 = {lw[1 * F * F + kk * F + nc], lw[1 * F * F + (kk + 1) * F + nc]};
            v2f b2 = {lw[2 * F * F + kk * F + nc], lw[2 * F * F + (kk + 1) * F + nc]};
            acch[t] = __builtin_amdgcn_wmma_f32_16x16x4_f32(false, ax, false, b0, (short)0, acch[t], false, false);
            acch[t] = __builtin_amdgcn_wmma_f32_16x16x4_f32(false, a1, false, b1, (short)0, acch[t], false, false);
            acch[t] = __builtin_amdgcn_wmma_f32_16x16x4_f32(false, a2, false, b2, (short)0, acch[t], false, false);
        }
    }

    // ---------- fused epilogue: z = sigmoid, h~ = tanh, h = (1-z)*h~ ----------
    if (valid) {
#pragma unroll
        for (int t = 0; t < 4; ++t) {
            const int nc = t * 16 + m;
            const float bzv = bxz[nc] + bhz[nc];
            const float bhv = bxh[nc] + bhh[nc];
#pragma unroll
            for (int r = 0; r < 8; ++r) {
                const int mm = r + 8 * kh;  // C/D layout: vgpr r, lanes16-31 -> M+8
                float z  = 1.f / (1.f + __expf(-(accz[t][r] + bzv)));
                float ht = tanhf(acch[t][r] + bhv);
                hout[(long)(baseRow + mm) * F + nc] = (1.f - z) * ht;
            }
        }
    }
}

// out[i] = sigmoid(tanh(h[i]) @ Wlin + blin), Wlin: [64][2]
__global__ void head_kernel(const float* __restrict__ h, const float* __restrict__ Wlin,
                            const float* __restrict__ blin, float* __restrict__ out, int n) {
    int i = blockIdx.x * blockDim.x + threadIdx.x;
    if (i >= n) return;
    float a0 = blin[0], a1 = blin[1];
    const float* hr = h + (long)i * F;
#pragma unroll
    for (int k = 0; k < F; ++k) {
        float t = tanhf(hr[k]);
        a0 += t * Wlin[k * 2 + 0];
        a1 += t * Wlin[k * 2 + 1];
    }
    out[i * 2 + 0] = 1.f / (1.f + __expf(-a0));
    out[i * 2 + 1] = 1.f / (1.f + __expf(-a1));
}

extern "C" void kernel_launch(void* const* d_in, const int* in_sizes, int n_in,
                              void* d_out, int out_size, void* d_ws, size_t ws_size,
                              hipStream_t stream) {
    const float* x    = (const float*)d_in[0];
    const int*   ei   = (const int*)  d_in[1];
    const float* w    = (const float*)d_in[2];
    const float* Wxz  = (const float*)d_in[3];
    const float* bxz  = (const float*)d_in[4];
    const float* bhz  = (const float*)d_in[6];
    const float* Wxh  = (const float*)d_in[11];
    const float* bxh  = (const float*)d_in[12];
    const float* bhh  = (const float*)d_in[14];
    const float* Wlin = (const float*)d_in[15];
    const float* blin = (const float*)d_in[16];
    // r-gate weights (d_in[7..10]) are dead: h0 == 0 in the reference.

    const int n = in_sizes[0] / F;     // 50000
    const int E = in_sizes[2];         // 800000
    const int* row = ei;
    const int* col = ei + E;

    float* ws  = (float*)d_ws;
    float* deg = ws;                           // n
    float* tx1 = deg + n;                      // n*F
    float* p2  = tx1 + (size_t)n * F;          // n*F  (reused as h: per-wave row ownership)
    float* nrm = p2  + (size_t)n * F;          // E
    float* h   = p2;

    const long nzero = (long)n + 2L * n * F;   // deg + tx1 + p2
    zero_f32<<<2048, 256, 0, stream>>>(ws, nzero);

    deg_kernel <<<(E + 255) / 256, 256, 0, stream>>>(row, w, deg, E);
    norm_kernel<<<(E + 255) / 256, 256, 0, stream>>>(row, col, w, deg, nrm, E);

    // tx1 = prop(x);  p2 = prop(tx1)   (tx2 = 2*p2 - x computed in-flight)
    prop_kernel<<<(E + 7) / 8, 256, 0, stream>>>(x,   tx1, row, col, nrm, E);
    prop_kernel<<<(E + 7) / 8, 256, 0, stream>>>(tx1, p2,  row, col, nrm, E);

    const int tiles = (n + 15) / 16;           // 3125 wave-tiles
    gemm_gate_kernel<<<(tiles + 7) / 8, 256, 0, stream>>>(
        x, tx1, p2, Wxz, Wxh, bxz, bhz, bxh, bhh, h, n);

    head_kernel<<<(n + 255) / 256, 256, 0, stream>>>(h, Wlin, blin, (float*)d_out, n);
}